// thetaRNNLayer_41437844472167
// MI455X (gfx1250) — compile-verified
//
#include <hip/hip_runtime.h>

// Problem constants (match reference)
#define SEQ    2048
#define HIDDEN 2048
#define INPUT  2048

typedef __attribute__((ext_vector_type(2))) float    v2f;
typedef __attribute__((ext_vector_type(8))) float    v8f;
typedef __attribute__((ext_vector_type(4))) unsigned u32x4;
typedef __attribute__((ext_vector_type(8))) int      i32x8;
typedef __attribute__((ext_vector_type(4))) int      i32x4;

// GEMM tiling
#define BM  128
#define BN  128
#define BK  16
#define LDP 20   // padded LDS row stride (words): 16B-aligned float4 stores,
                 // 20*m mod 64 distinct for m=0..15 -> conflict-free fragments

// ---------------------------------------------------------------------------
// LDS-tiled WMMA f32 GEMM:
//   D[s,h] = opt_relu( A[s,:] . W[h,:] + add[s,h] + bias[h] )
// Block: 256 threads / 8 waves, 128x128 output tile, K-step 16.
// Each wave owns a 16x128 strip = 8 accumulators (v8f each).
// Register-prefetch pipelining: next K-panel loads issue before compute.
// Fragment layout per CDNA5 ISA 7.12.2 (f32 16x16x4):
//   A: lane<16 -> M=lane, K=kb+{0,1}, kb=0; lane>=16 -> M=lane-16, kb=2.
//   B: symmetric (N = lane&15).  C/D: VGPR v -> M=v+8*(lane>>4), N=lane&15.
// ---------------------------------------------------------------------------
__global__ __launch_bounds__(256)
void wmma_gemm_f32(const float* __restrict__ A,     // [S,K] row-major
                   const float* __restrict__ W,     // [H,K] row-major (B = W^T)
                   const float* __restrict__ add,   // [S,H]
                   const float* __restrict__ bias,  // [H]
                   float* __restrict__ D,           // [S,H]
                   int S, int K, int H, int do_relu)
{
    __shared__ float Alds[BM * LDP];
    __shared__ float Wlds[BN * LDP];

    const int tid    = threadIdx.x;
    const int lane   = tid & 31;
    const int wv     = tid >> 5;            // wave 0..7 -> M strip
    const int tilesN = H / BN;
    const int tm     = blockIdx.x / tilesN;
    const int tn     = blockIdx.x % tilesN;

    const int mrow = lane & 15;
    const int kb   = (lane >> 4) << 1;      // 0 or 2

    // Global staging: 512 float4 per 128x16 panel; thread covers j = tid, tid+256
    const int r0 = tid >> 2;                // 0..63
    const int c0 = (tid & 3) << 2;          // 0,4,8,12
    const int r1 = r0 + 64;

    const float* __restrict__ Abase = A + (size_t)(tm * BM) * K;
    const float* __restrict__ Wbase = W + (size_t)(tn * BN) * K;

    v8f acc[8] = {};
    const int nK = K / BK;

    float4 pa0 = *(const float4*)(Abase + (size_t)r0 * K + c0);
    float4 pa1 = *(const float4*)(Abase + (size_t)r1 * K + c0);
    float4 pw0 = *(const float4*)(Wbase + (size_t)r0 * K + c0);
    float4 pw1 = *(const float4*)(Wbase + (size_t)r1 * K + c0);

    for (int ks = 0; ks < nK; ++ks) {
        *(float4*)&Alds[r0 * LDP + c0] = pa0;
        *(float4*)&Alds[r1 * LDP + c0] = pa1;
        *(float4*)&Wlds[r0 * LDP + c0] = pw0;
        *(float4*)&Wlds[r1 * LDP + c0] = pw1;
        __syncthreads();

        if (ks + 1 < nK) {                  // prefetch next K panel into regs
            const float* An = Abase + (size_t)(ks + 1) * BK;
            const float* Wn = Wbase + (size_t)(ks + 1) * BK;
            pa0 = *(const float4*)(An + (size_t)r0 * K + c0);
            pa1 = *(const float4*)(An + (size_t)r1 * K + c0);
            pw0 = *(const float4*)(Wn + (size_t)r0 * K + c0);
            pw1 = *(const float4*)(Wn + (size_t)r1 * K + c0);
        }

        const float* Af = &Alds[(wv * 16 + mrow) * LDP + kb];
#pragma unroll
        for (int kk = 0; kk < BK; kk += 4) {
            v2f a = *(const v2f*)(Af + kk);
#pragma unroll
            for (int nt = 0; nt < 8; ++nt) {
                v2f b = *(const v2f*)&Wlds[(nt * 16 + mrow) * LDP + kk + kb];
                acc[nt] = __builtin_amdgcn_wmma_f32_16x16x4_f32(
                    false, a, false, b, (short)0, acc[nt], false, false);
            }
        }
        __syncthreads();
    }

    // Epilogue: + internal + bias, optional relu
    const int mbase = tm * BM + wv * 16 + ((lane >> 4) << 3);
#pragma unroll
    for (int nt = 0; nt < 8; ++nt) {
        const int n  = tn * BN + nt * 16 + (lane & 15);
        const float bv = bias[n];
#pragma unroll
        for (int v = 0; v < 8; ++v) {
            size_t idx = (size_t)(mbase + v) * H + n;
            float val = acc[nt][v] + add[idx] + bv;
            if (do_relu) val = fmaxf(val, 0.0f);
            D[idx] = val;
        }
    }
}

// ---------------------------------------------------------------------------
// Sequential h0 chain: h0[t] = relu(P0[t] + Whh . h0[t-1]),  h0[-1] = state.
// ---------------------------------------------------------------------------
#define CHAIN_BLOCKS  16
#define CHAIN_THREADS 256

__global__ void h0_chain_kernel(const float* __restrict__ P0,
                                const float* __restrict__ Whh,
                                const float* __restrict__ state,
                                float* __restrict__ H0,
                                float* __restrict__ hfinal,
                                unsigned* __restrict__ cnt,
                                int S, int Hd)
{
    __shared__ float sh[HIDDEN];
    const int tid   = threadIdx.x;
    const int gtid  = blockIdx.x * blockDim.x + tid;
    const int row   = gtid >> 1;
    const int half  = gtid & 1;
    const int halfK = Hd >> 1;
    const int nq    = halfK >> 2;

    const float4* __restrict__ wrow =
        (const float4*)(Whh + (size_t)row * Hd + half * halfK);
    const unsigned nb = gridDim.x;

    for (int t = 0; t < S; ++t) {
        const float* hprev = (t == 0) ? state : (H0 + (size_t)(t - 1) * Hd);
        for (int i = tid; i < (Hd >> 2); i += blockDim.x)
            ((float4*)sh)[i] = ((const float4*)hprev)[i];
        __syncthreads();

        const float4* hv = (const float4*)(sh + half * halfK);
        float sum = 0.0f;
#pragma unroll 4
        for (int i = 0; i < nq; ++i) {
            float4 w = wrow[i];
            float4 h = hv[i];
            sum += w.x * h.x + w.y * h.y + w.z * h.z + w.w * h.w;
        }
        sum += __shfl_xor(sum, 1, 32);

        if (half == 0) {
            float v = P0[(size_t)t * Hd + row] + sum;
            v = fmaxf(v, 0.0f);
            H0[(size_t)t * Hd + row] = v;
            if (t == S - 1) hfinal[row] = v;
        }

        __threadfence();
        __syncthreads();
        if (tid == 0) {
            atomicAdd(cnt, 1u);
            const unsigned target = nb * (unsigned)(t + 1);
            while (atomicAdd(cnt, 0u) < target)
                __builtin_amdgcn_s_sleep(1);
        }
        __syncthreads();
        __threadfence();
    }
}

__global__ void init_cnt_kernel(unsigned* cnt)
{
    if (threadIdx.x == 0 && blockIdx.x == 0) *cnt = 0u;
}

// ---------------------------------------------------------------------------
// TDM probe (compiled, NOT launched): stages a 16x128 f32 tile from `src`
// into LDS via the Tensor Data Mover, per CDNA5 ISA §8 D# layout.
// Group0: count=1, lds_addr, global_addr[56:0], type=2.
// Group1: data_size=2 (4B), tensor_dim0/1=2048, tile_dim0=128, tile_dim1=16,
//         tensor_dim0_stride=2048.  Groups 2/3/4 zero (2-D tensor).
// This toolchain (clang-23 / therock-10.0) uses the 6-arg builtin form:
//   (uint32x4, int32x8, int32x4, int32x4, int32x8, i32 cpol)
// Exercises: tensor_load_to_lds, s_wait_tensorcnt, cluster id / barrier.
// ---------------------------------------------------------------------------
__global__ void tdm_tensor_probe(const float* __restrict__ src,
                                 float* __restrict__ dst)
{
    __shared__ float tile[BK * BN];

    if (threadIdx.x == 0) {
        unsigned lds_addr = (unsigned)(size_t)&tile[0];
        unsigned long long ga = (unsigned long long)(size_t)src;

        u32x4 g0;
        g0[0] = 1u;                                           // count=1 (valid)
        g0[1] = lds_addr;                                     // lds_addr
        g0[2] = (unsigned)(ga & 0xFFFFFFFFu);                 // global_addr[31:0]
        g0[3] = (unsigned)((ga >> 32) & 0x01FFFFFFu)          // global_addr[56:32]
              | (2u << 30);                                   // type=2 ("image")

        i32x8 g1;
        g1[0] = (int)(2u << 16);                              // data_size=4B
        g1[1] = (int)(2048u << 16);                           // tensor_dim0 lo16
        g1[2] = (int)((2048u >> 16) | (2048u << 16));         // dim0 hi | dim1 lo
        g1[3] = (int)((2048u >> 16) | (128u << 16));          // dim1 hi | tile_dim0
        g1[4] = 16;                                           // tile_dim1 (tile_dim2=0)
        g1[5] = 2048;                                         // tensor_dim0_stride lo32
        g1[6] = 0;                                            // stride0 hi | stride1 lo
        g1[7] = 0;

        i32x4 gz4 = {0, 0, 0, 0};
        i32x8 gz8 = {0, 0, 0, 0, 0, 0, 0, 0};
        __builtin_amdgcn_tensor_load_to_lds(g0, g1, gz4, gz4, gz8, 0);
        __builtin_amdgcn_s_wait_tensorcnt(0);
    }
    __syncthreads();

    int cid = __builtin_amdgcn_cluster_id_x();                // 0 if not clustered
    __builtin_amdgcn_s_cluster_barrier();                     // NOP outside cluster
    dst[threadIdx.x + (unsigned)cid] = tile[threadIdx.x];
}

// ---------------------------------------------------------------------------
// Orchestration.
//   d_in: 0=input [1,S,I]  1=internal [4,S,H]  2=state [1,1,H]
//         3=weight_ih [H,I]  4=weight_hh [H,H]  5=bias [H]  6=theta (==3)
//   d_out: outputs [4,S,H] then h_final [H]
//   P0 staged in d_out segment 3 (dead before the theta=3 GEMM overwrites it).
// ---------------------------------------------------------------------------
extern "C" void kernel_launch(void* const* d_in, const int* in_sizes, int n_in,
                              void* d_out, int out_size, void* d_ws, size_t ws_size,
                              hipStream_t stream)
{
    const float* input    = (const float*)d_in[0];
    const float* internal = (const float*)d_in[1];
    const float* state    = (const float*)d_in[2];
    const float* w_ih     = (const float*)d_in[3];
    const float* w_hh     = (const float*)d_in[4];
    const float* bias     = (const float*)d_in[5];

    float* out = (float*)d_out;
    const size_t segN = (size_t)SEQ * HIDDEN;
    float* seg0   = out;             // H0
    float* seg1   = out + 1 * segN;  // H1
    float* seg2   = out + 2 * segN;  // H2
    float* seg3   = out + 3 * segN;  // H3 (temporarily holds P0)
    float* hfinal = out + 4 * segN;

    unsigned* cnt = (unsigned*)d_ws;

    const int blocks = (SEQ / BM) * (HIDDEN / BN);   // 256 blocks of 128x128 tiles

    // 0) zero the barrier counter (deterministic per launch, graph-safe)
    init_cnt_kernel<<<1, 64, 0, stream>>>(cnt);

    // 1) P0 = input @ Wih^T + internal[0] + bias (no relu) -> seg3 scratch
    wmma_gemm_f32<<<blocks, 256, 0, stream>>>(input, w_ih, internal, bias,
                                              seg3, SEQ, INPUT, HIDDEN, 0);

    // 2) sequential h0 chain -> seg0, h_final
    h0_chain_kernel<<<CHAIN_BLOCKS, CHAIN_THREADS, 0, stream>>>(
        seg3, w_hh, state, seg0, hfinal, cnt, SEQ, HIDDEN);

    // 3) H_th = relu(internal[th] + bias + H_{th-1} @ Whh^T), th = 1..3
    wmma_gemm_f32<<<blocks, 256, 0, stream>>>(seg0, w_hh, internal + 1 * segN,
                                              bias, seg1, SEQ, HIDDEN, HIDDEN, 1);
    wmma_gemm_f32<<<blocks, 256, 0, stream>>>(seg1, w_hh, internal + 2 * segN,
                                              bias, seg2, SEQ, HIDDEN, HIDDEN, 1);
    wmma_gemm_f32<<<blocks, 256, 0, stream>>>(seg2, w_hh, internal + 3 * segN,
                                              bias, seg3, SEQ, HIDDEN, HIDDEN, 1);
}